// YOLOV3TargetMerger_84275848282254
// MI455X (gfx1250) — compile-verified
//
#include <hip/hip_runtime.h>
#include <hip/hip_bf16.h>
#include <stdint.h>

#define IGNORE_IOU_THRESH 0.7f
#define IOU_EPS 1e-12f
#define M_GT 50
#define C_CLS 80
#define BATCH 32

typedef float v4f __attribute__((ext_vector_type(4)));
typedef float v2f __attribute__((ext_vector_type(2)));
typedef unsigned int v4u __attribute__((ext_vector_type(4)));
typedef int v4i __attribute__((ext_vector_type(4)));
typedef int v8i __attribute__((ext_vector_type(8)));

#if __has_builtin(__builtin_amdgcn_tensor_load_to_lds)
#define HAVE_TDM 1
#else
#define HAVE_TDM 0
#endif

// ---- LDS byte-offset of a __shared__ object (generic -> addrspace(3) -> int) ----
__device__ __forceinline__ uint32_t lds_byte_offset(const void* p) {
  return (uint32_t)(uintptr_t)(__attribute__((address_space(3))) const void*)p;
}

__device__ __forceinline__ void wait_tensorcnt0() {
#if __has_builtin(__builtin_amdgcn_s_wait_tensorcnt)
  __builtin_amdgcn_s_wait_tensorcnt(0);
#else
  asm volatile("s_wait_tensorcnt 0x0" ::: "memory");
#endif
}

#if HAVE_TDM
// Stage `nfloats` contiguous floats from global memory into LDS via the
// Tensor Data Mover (1-D tile descriptor, data_size = 4 bytes).
__device__ __forceinline__ void tdm_load_floats_to_lds(const float* gsrc,
                                                       void* lds_dst,
                                                       uint32_t nfloats) {
  const uint64_t ga = (uint64_t)(uintptr_t)gsrc;
  const uint32_t la = lds_byte_offset(lds_dst);

  // D# group 0 (128b): count=1 | lds_addr | global_addr[56:0] | type=2
  v4u g0;
  g0.x = 0x1u;                                   // count = 1 valid descriptor
  g0.y = la;                                     // LDS byte address
  g0.z = (uint32_t)ga;                           // global addr [31:0]
  g0.w = (uint32_t)(ga >> 32) | (2u << 30);      // global addr [56:32], type=2

  // D# group 1 (256b)
  v8i g1;
  g1[0] = (int)(2u << 16);                       // data_size = 2 (4 bytes), no mcast
  g1[1] = (int)(nfloats << 16);                  // tensor_dim0[15:0] in bits [31:16]
  g1[2] = (int)((nfloats >> 16) | (1u << 16));   // tensor_dim0[31:16], tensor_dim1 = 1
  g1[3] = (int)(nfloats << 16);                  // tile_dim0 in bits [31:16]
  g1[4] = 0;                                     // tile_dim1 = 0 (unused), tile_dim2 = 0
  g1[5] = (int)nfloats;                          // tensor_dim0_stride[31:0]
  g1[6] = 0;                                     // stride0 hi, stride1 lo
  g1[7] = 0;                                     // stride1 hi

  v4i g2 = {0, 0, 0, 0};
  v4i g3 = {0, 0, 0, 0};
#if defined(__clang_major__) && __clang_major__ >= 23
  v8i g4 = {0, 0, 0, 0, 0, 0, 0, 0};
  __builtin_amdgcn_tensor_load_to_lds(g0, g1, g2, g3, g4, 0);
#else
  __builtin_amdgcn_tensor_load_to_lds(g0, g1, g2, g3, 0);
#endif
}
#endif

// ---------------------------------------------------------------------------
// Kernel 1: per-anchor pass. One thread per (batch, anchor n).
//   - GT boxes (50 x 4 floats = 800 B) staged into LDS by the TDM.
//   - max IoU over 50 GT boxes, dyn_obj = -(iou > 0.7)
//   - masked merge of objectness / centers / scales / weights.
// Streaming outputs use non-temporal stores (write-once data).
// ---------------------------------------------------------------------------
__global__ void __launch_bounds__(256)
yolo_merge_anchor_kernel(const float* __restrict__ box_preds,
                         const float* __restrict__ gt_boxes,
                         const float* __restrict__ obj_t,
                         const float* __restrict__ centers_t,
                         const float* __restrict__ scales_t,
                         const float* __restrict__ weights_t,
                         float* __restrict__ out,
                         int N, long long BN) {
  __shared__ float sgt[M_GT * 4];
  const int bidx = blockIdx.y;
  const float* gsrc = gt_boxes + (size_t)bidx * (M_GT * 4);

#if HAVE_TDM
  if (threadIdx.x < 32u) {          // wave 0 only: issue TDM, wait for it
    tdm_load_floats_to_lds(gsrc, sgt, M_GT * 4);
    wait_tensorcnt0();
  }
#else
  for (int i = threadIdx.x; i < M_GT * 4; i += blockDim.x) sgt[i] = gsrc[i];
#endif
  __syncthreads();

  const int n = blockIdx.x * blockDim.x + threadIdx.x;
  if (n >= N) return;
  const long long row = (long long)bidx * N + n;

  const v4f bp = ((const v4f*)box_preds)[row];
  const float area_p = (bp.z - bp.x) * (bp.w - bp.y);

  float best = -1.0f;
#pragma unroll 5
  for (int m = 0; m < M_GT; ++m) {
    const v4f g = *((const v4f*)(sgt + m * 4));   // contiguous -> ds_load_b128
    const float tlx = fmaxf(bp.x, g.x);
    const float tly = fmaxf(bp.y, g.y);
    const float brx = fminf(bp.z, g.z);
    const float bry = fminf(bp.w, g.w);
    const float iw = fmaxf(brx - tlx, 0.0f);
    const float ih = fmaxf(bry - tly, 0.0f);
    const float inter = iw * ih;
    const float area_g = (g.z - g.x) * (g.w - g.y);
    const float iou = inter / (area_p + area_g - inter + IOU_EPS);
    best = fmaxf(best, iou);
  }
  const float dyn_obj = (best > IGNORE_IOU_THRESH) ? -1.0f : 0.0f;

  const float obj = obj_t[row];
  const bool mask = obj > 0.0f;

  __builtin_nontemporal_store(mask ? obj : dyn_obj, out + row);

  const v2f c = __builtin_nontemporal_load((const v2f*)centers_t + row);
  const v2f s = __builtin_nontemporal_load((const v2f*)scales_t + row);
  const v2f w = __builtin_nontemporal_load((const v2f*)weights_t + row);
  const v2f z = {0.0f, 0.0f};
  __builtin_nontemporal_store(mask ? c : z, (v2f*)(out + 1 * BN) + row);
  __builtin_nontemporal_store(mask ? s : z, (v2f*)(out + 3 * BN) + row);
  __builtin_nontemporal_store(mask ? w : z, (v2f*)(out + 5 * BN) + row);
}

// ---------------------------------------------------------------------------
// Kernel 2: class-target merge. Pure streaming pass over b*N*C elements in
// float4 vectors (C = 80, divisible by 4 -> a vector never crosses a row).
//   class_targets = mask ? clas : -1
//   class_mask    = (mask && clas >= 0) ? 1 : 0
// 233 MB read-once + 466 MB write-once -> non-temporal on both streams;
// obj_t (2.9 MB, reused 20x per row) stays regular so it lives in L2.
// ---------------------------------------------------------------------------
__global__ void __launch_bounds__(256)
yolo_merge_class_kernel(const float* __restrict__ obj_t,
                        const float* __restrict__ clas_t,
                        float* __restrict__ out_ct,
                        float* __restrict__ out_cm,
                        long long nvec) {
  const long long i = (long long)blockIdx.x * blockDim.x + threadIdx.x;
  if (i >= nvec) return;

  const long long row = (i * 4) / C_CLS;
  const bool mask = obj_t[row] > 0.0f;

  const v4f v = __builtin_nontemporal_load((const v4f*)clas_t + i);
  v4f ct, cm;
  ct.x = mask ? v.x : -1.0f;  cm.x = (mask && v.x >= 0.0f) ? 1.0f : 0.0f;
  ct.y = mask ? v.y : -1.0f;  cm.y = (mask && v.y >= 0.0f) ? 1.0f : 0.0f;
  ct.z = mask ? v.z : -1.0f;  cm.z = (mask && v.z >= 0.0f) ? 1.0f : 0.0f;
  ct.w = mask ? v.w : -1.0f;  cm.w = (mask && v.w >= 0.0f) ? 1.0f : 0.0f;

  __builtin_nontemporal_store(ct, (v4f*)out_ct + i);
  __builtin_nontemporal_store(cm, (v4f*)out_cm + i);
}

extern "C" void kernel_launch(void* const* d_in, const int* in_sizes, int n_in,
                              void* d_out, int out_size, void* d_ws, size_t ws_size,
                              hipStream_t stream) {
  const float* box_preds = (const float*)d_in[0];
  const float* gt_boxes  = (const float*)d_in[1];
  const float* obj_t     = (const float*)d_in[2];
  const float* centers_t = (const float*)d_in[3];
  const float* scales_t  = (const float*)d_in[4];
  const float* weights_t = (const float*)d_in[5];
  const float* clas_t    = (const float*)d_in[6];
  float* out = (float*)d_out;

  const int b = BATCH;
  const int N = in_sizes[2] / b;                // obj_t holds b*N elements
  const long long BN = (long long)b * N;

  // Output layout (floats, concatenated in reference return order):
  //   [0,BN)        objectness
  //   [BN,3BN)      center_targets
  //   [3BN,5BN)     scale_targets
  //   [5BN,7BN)     weights
  //   [7BN,87BN)    class_targets
  //   [87BN,167BN)  class_mask
  dim3 blk1(256);
  dim3 grd1((N + 255) / 256, b);
  yolo_merge_anchor_kernel<<<grd1, blk1, 0, stream>>>(
      box_preds, gt_boxes, obj_t, centers_t, scales_t, weights_t, out, N, BN);

  const long long nvec = BN * C_CLS / 4;
  const int blocks2 = (int)((nvec + 255) / 256);
  yolo_merge_class_kernel<<<blocks2, 256, 0, stream>>>(
      obj_t, clas_t, out + 7 * BN, out + 87 * BN, nvec);
}